// Until_15753940041804
// MI455X (gfx1250) — compile-verified
//
#include <hip/hip_runtime.h>
#include <stdint.h>

// Problem constants (match reference: B=64, T=2048, C=2, window 64)
#define T_DIM 2048
#define TILE  128
#define WIN   64
#define NLD   (2 * (TILE + WIN))   // 384 interleaved floats per staged array

// Low 32 bits of a flat shared-aperture address are the LDS byte offset
// (see CDNA5 ISA); async-to-LDS VDST takes that offset.
__device__ __forceinline__ unsigned lds_off_u32(const void* p) {
  return (unsigned)(uintptr_t)p;
}

__global__ __launch_bounds__(256) void softwin_kernel(
    const float* __restrict__ phi, const float* __restrict__ psi,
    const int* __restrict__ scale, float* __restrict__ out)
{
  __shared__ float lds_p[NLD];
  __shared__ float lds_q[NLD];

  const int tid = threadIdx.x;
  const int t0  = blockIdx.x * TILE;
  const int b   = blockIdx.y;
  const float s = (float)scale[0];

  const size_t rowbase = ((size_t)b * T_DIM + t0) * 2;  // element offset of tile start
  const float* gphi = phi + rowbase;                    // uniform -> SGPR pair
  const float* gpsi = psi + rowbase;

  // clamp so the last tile never reads past the end of the batch row
  const int iMax = 2 * (T_DIM - t0) - 1;

  const unsigned ldsP = lds_off_u32(lds_p);
  const unsigned ldsQ = lds_off_u32(lds_q);

  // --- async stage: global -> LDS (ASYNCcnt path, gfx1250) -------------------
  for (int i = tid; i < NLD; i += 256) {
    const int ie = (i < iMax) ? i : iMax;
    const unsigned voff = (unsigned)(ie * 4);
    const unsigned dP = ldsP + (unsigned)(i * 4);
    const unsigned dQ = ldsQ + (unsigned)(i * 4);
    asm volatile("global_load_async_to_lds_b32 %0, %1, %2"
                 :: "v"(dP), "v"(voff), "s"(gphi) : "memory");
    asm volatile("global_load_async_to_lds_b32 %0, %1, %2"
                 :: "v"(dQ), "v"(voff), "s"(gpsi) : "memory");
  }
  asm volatile("s_wait_asynccnt 0x0" ::: "memory");
  __syncthreads();

  // --- convert in place: p = exp(-s*phi), q = exp(-s*psi) --------------------
  const float nsl = -s * 1.4426950408889634f;  // -s * log2(e)
  for (int i = tid; i < NLD; i += 256) {
    lds_p[i] = __builtin_amdgcn_exp2f(nsl * lds_p[i]);
    lds_q[i] = __builtin_amdgcn_exp2f(nsl * lds_q[i]);
  }
  __syncthreads();

  // --- window reduction: best = log( sum_k 1/(E_k + q) ) / s -----------------
  // thread tid -> c = tid&1, t = t0 + (tid>>1); LDS index at step k is tid+2k
  const int t = t0 + (tid >> 1);
  const float E0 = __builtin_amdgcn_exp2f(nsl);            // exp(-s * 1.0), k==0 slot
  float acc = __builtin_amdgcn_rcpf(E0 + lds_q[tid]);
  float P   = lds_p[tid];

  if (t + WIN <= T_DIM) {
    #pragma unroll
    for (int k = 1; k < WIN; ++k) {
      acc += __builtin_amdgcn_rcpf(P + lds_q[tid + 2 * k]);
      P   += lds_p[tid + 2 * k];
    }
  } else {
    const int K = T_DIM - t;   // >= 1
    for (int k = 1; k < K; ++k) {
      acc += __builtin_amdgcn_rcpf(P + lds_q[tid + 2 * k]);
      P   += lds_p[tid + 2 * k];
    }
  }

  out[rowbase + tid] = __builtin_amdgcn_logf(acc) * (0.6931471805599453f / s);
}

extern "C" void kernel_launch(void* const* d_in, const int* in_sizes, int n_in,
                              void* d_out, int out_size, void* d_ws, size_t ws_size,
                              hipStream_t stream) {
  (void)n_in; (void)out_size; (void)d_ws; (void)ws_size;
  const float* phi   = (const float*)d_in[0];
  const float* psi   = (const float*)d_in[1];
  const int*   scale = (const int*)d_in[2];
  float* out = (float*)d_out;

  const int B = in_sizes[0] / (T_DIM * 2);   // 64 for the reference shapes
  dim3 grid(T_DIM / TILE, B);
  softwin_kernel<<<grid, 256, 0, stream>>>(phi, psi, scale, out);
}